// FNSAttention_22162031247484
// MI455X (gfx1250) — compile-verified
//
#include <hip/hip_runtime.h>
#include <hip/hip_bf16.h>
#include <math.h>

// ---------------------------------------------------------------------------
// FNS (geodesic) attention, MI455X / gfx1250.
// B=2, N=256, E=768, H=12, D=64.  All f32 (WMMA F32_16X16X4_F32 keeps the
// reference precision; GEMMs are tiny, APSP-in-LDS dominates).
// ---------------------------------------------------------------------------

#define Bsz 2
#define Nsz 256
#define Esz 768
#define Hsz 12
#define Dsz 64
#define BH  (Bsz * Hsz)          // 24
#define BN  (Bsz * Nsz)          // 512
#define INF_DIST 1e9f

typedef __attribute__((ext_vector_type(2))) float v2f;
typedef __attribute__((ext_vector_type(8))) float v8f;

__device__ __forceinline__ v8f wmma_f32_k4(v2f a, v2f b, v8f c) {
  // D = A(16x4 f32) * B(4x16 f32) + C(16x16 f32)
  return __builtin_amdgcn_wmma_f32_16x16x4_f32(
      /*neg_a=*/false, a, /*neg_b=*/false, b,
      /*c_mod=*/(short)0, c, /*reuse_a=*/false, /*reuse_b=*/false);
}

// ---------------------------------------------------------------------------
// Kernel 1: fused Q/K/V projection.  X[512,768] @ W[768,768] + bias,
// stored as [B,H,N,D].  One wave per 16x16 output tile.
// waves = 3 * (512/16) * (768/16) = 4608
// ---------------------------------------------------------------------------
__global__ void __launch_bounds__(128) qkv_gemm_kernel(
    const float* __restrict__ X,
    const float* __restrict__ Wq, const float* __restrict__ bq,
    const float* __restrict__ Wk, const float* __restrict__ bk,
    const float* __restrict__ Wv, const float* __restrict__ bv,
    float* __restrict__ q, float* __restrict__ k, float* __restrict__ v) {
  const int wid  = (blockIdx.x * blockDim.x + threadIdx.x) >> 5;
  const int lane = threadIdx.x & 31;
  if (wid >= 3 * 32 * 48) return;
  const int p   = wid / (32 * 48);
  const int r   = wid % (32 * 48);
  const int tm  = r / 48;
  const int tn  = r % 48;
  const float* W    = (p == 0) ? Wq : (p == 1) ? Wk : Wv;
  const float* bias = (p == 0) ? bq : (p == 1) ? bk : bv;
  float* Out        = (p == 0) ? q  : (p == 1) ? k  : v;

  const int arow = tm * 16 + (lane & 15);    // A-matrix row (bn index)
  const int koff = (lane >> 4) * 2;          // A/B K sub-offset per lane half
  const int ncol = tn * 16 + (lane & 15);    // B-matrix / output column (e)

  v8f c = {};
  for (int kk = 0; kk < Esz; kk += 4) {
    v2f a, b;
    const float* ap = X + arow * Esz + kk + koff;
    a.x = ap[0];
    a.y = ap[1];
    b.x = W[(kk + koff) * Esz + ncol];
    b.y = W[(kk + koff + 1) * Esz + ncol];
    c = wmma_f32_k4(a, b, c);
  }

  const float bb = bias[ncol];
  const int h = ncol >> 6;          // head
  const int d = ncol & 63;          // dim in head
  const int mbase = tm * 16 + ((lane >> 4) << 3);
  for (int vv = 0; vv < 8; ++vv) {
    const int bn = mbase + vv;            // 0..511
    const int b_ = bn >> 8;
    const int n  = bn & 255;
    Out[((b_ * Hsz + h) * Nsz + n) * Dsz + d] = c[vv] + bb;
  }
}

// ---------------------------------------------------------------------------
// Kernel 2: row squared-norms of q and k.  One wave per row (12288 rows).
// ---------------------------------------------------------------------------
__global__ void __launch_bounds__(256) rownorm_kernel(
    const float* __restrict__ q, const float* __restrict__ k,
    float* __restrict__ q2, float* __restrict__ k2) {
  const int wid  = (blockIdx.x * blockDim.x + threadIdx.x) >> 5;
  const int lane = threadIdx.x & 31;
  if (wid >= 2 * BH * Nsz) return;
  const float* src = (wid < BH * Nsz) ? q : k;
  float* dst       = (wid < BH * Nsz) ? q2 : k2;
  const int rowi = wid % (BH * Nsz);
  const float* r = src + rowi * Dsz;
  float a0 = r[lane], a1 = r[lane + 32];
  float s = a0 * a0 + a1 * a1;
  for (int off = 16; off; off >>= 1) s += __shfl_xor(s, off, 32);
  if (lane == 0) dst[rowi] = s;
}

// ---------------------------------------------------------------------------
// Kernel 3: cdist + mask.  Per (b,h): dist = sqrt(max(q2_i + k2_j - 2 QK^T,
// 1e-12)), INF where masked.  One wave per 16x16 tile; waves = 24*16*16.
// ---------------------------------------------------------------------------
__global__ void __launch_bounds__(128) cdist_kernel(
    const float* __restrict__ q, const float* __restrict__ k,
    const float* __restrict__ q2, const float* __restrict__ k2,
    const float* __restrict__ amask, float* __restrict__ dist) {
  const int wid  = (blockIdx.x * blockDim.x + threadIdx.x) >> 5;
  const int lane = threadIdx.x & 31;
  if (wid >= BH * 16 * 16) return;
  const int bh = wid / 256;
  const int r  = wid % 256;
  const int ti = r / 16;
  const int tj = r % 16;

  const int arow = ti * 16 + (lane & 15);   // i
  const int jcol = tj * 16 + (lane & 15);   // j
  const int koff = (lane >> 4) * 2;
  const float* qb = q + bh * Nsz * Dsz;
  const float* kb = k + bh * Nsz * Dsz;

  v8f c = {};
  for (int dd = 0; dd < Dsz; dd += 4) {
    v2f a, b;
    const float* ap = qb + arow * Dsz + dd + koff;
    a.x = ap[0];
    a.y = ap[1];
    const float* bp = kb + jcol * Dsz + dd + koff;  // B[K=d][N=j] = k[j][d]
    b.x = bp[0];
    b.y = bp[1];
    c = wmma_f32_k4(a, b, c);
  }

  const int b_  = bh / Hsz;
  const float mj  = amask[b_ * Nsz + jcol];
  const float k2j = k2[bh * Nsz + jcol];
  const int ibase = ti * 16 + ((lane >> 4) << 3);
  for (int vv = 0; vv < 8; ++vv) {
    const int i = ibase + vv;
    float sq = q2[bh * Nsz + i] + k2j - 2.0f * c[vv];
    float dg = sqrtf(fmaxf(sq, 1e-12f));
    const float mi = amask[b_ * Nsz + i];
    if (!(mi >= 0.0f && mj >= 0.0f)) dg = INF_DIST;
    dist[(bh * Nsz + i) * Nsz + jcol] = dg;
  }
}

// ---------------------------------------------------------------------------
// Kernel 4: Floyd-Warshall APSP, matrix resident in 256 KB of LDS (CDNA5:
// 320 KB/WGP).  One 1024-thread workgroup per (b,h).  Thread t owns a
// 16-row x 4-col block: cg=t&63 (j0=cg*4), rg=t>>6 (i0=rg*16).
// ds_*_b128 accesses hit each of the 64 banks exactly twice per wave ->
// conflict-free.  Row/col k are invariant at pivot k (non-negative diag),
// so one barrier per pivot suffices.
// ---------------------------------------------------------------------------
__global__ void __launch_bounds__(1024) fw_apsp_kernel(float* __restrict__ dist) {
  extern __shared__ float S[];              // 256*256 floats = 256 KB
  const int bh = blockIdx.x;
  float* G = dist + bh * Nsz * Nsz;
  const int t = threadIdx.x;

  // Load matrix into LDS (coalesced float4).
  for (int rep = 0; rep < 16; ++rep) {
    const int idx = rep * 4096 + t * 4;
    *(float4*)(S + idx) = *(const float4*)(G + idx);
  }
  __syncthreads();

  const int j0 = (t & 63) * 4;
  const int i0 = (t >> 6) * 16;

  for (int kk = 0; kk < Nsz; ++kk) {
    const float4 rowv = *(const float4*)(S + kk * Nsz + j0);   // D[k][j0..j0+3]
#pragma unroll
    for (int rr = 0; rr < 16; ++rr) {
      const int i = i0 + rr;
      const float cik = S[i * Nsz + kk];                       // broadcast read
      float4 own = *(const float4*)(S + i * Nsz + j0);
      own.x = fminf(own.x, cik + rowv.x);
      own.y = fminf(own.y, cik + rowv.y);
      own.z = fminf(own.z, cik + rowv.z);
      own.w = fminf(own.w, cik + rowv.w);
      *(float4*)(S + i * Nsz + j0) = own;
    }
    __syncthreads();
  }

  // Write geodesic distances back.
  for (int rep = 0; rep < 16; ++rep) {
    const int idx = rep * 4096 + t * 4;
    *(float4*)(G + idx) = *(const float4*)(S + idx);
  }
}

// ---------------------------------------------------------------------------
// Kernel 5: score transform + row softmax, in place on dist.
// score = (1 + g/10)^(-64.5); softmax over j.  One wave per row (6144 rows).
// ---------------------------------------------------------------------------
__global__ void __launch_bounds__(256) score_softmax_kernel(float* __restrict__ dist) {
  const int wid  = (blockIdx.x * blockDim.x + threadIdx.x) >> 5;
  const int lane = threadIdx.x & 31;
  if (wid >= BH * Nsz) return;
  float* row = dist + wid * Nsz;

  float s[8];
  float mx = -1e30f;
#pragma unroll
  for (int m = 0; m < 8; ++m) {
    const float g = row[lane + m * 32];
    const float sc = __expf(-64.5f * __logf(1.0f + 0.1f * g));
    s[m] = sc;
    mx = fmaxf(mx, sc);
  }
  for (int off = 16; off; off >>= 1) mx = fmaxf(mx, __shfl_xor(mx, off, 32));
  float sum = 0.0f;
#pragma unroll
  for (int m = 0; m < 8; ++m) {
    s[m] = __expf(s[m] - mx);
    sum += s[m];
  }
  for (int off = 16; off; off >>= 1) sum += __shfl_xor(sum, off, 32);
  const float inv = 1.0f / sum;
#pragma unroll
  for (int m = 0; m < 8; ++m) row[lane + m * 32] = s[m] * inv;
}

// ---------------------------------------------------------------------------
// Kernel 6: attn_out = attn_w @ V   per (b,h): [256,256]x[256,64].
// Stored flat [B,H,N,D] == flat [B,N,E] (the reference's raw reshape).
// One wave per 16x16 tile; waves = 24 * 16 * 4.
// ---------------------------------------------------------------------------
__global__ void __launch_bounds__(128) av_gemm_kernel(
    const float* __restrict__ attn, const float* __restrict__ v,
    float* __restrict__ out) {
  const int wid  = (blockIdx.x * blockDim.x + threadIdx.x) >> 5;
  const int lane = threadIdx.x & 31;
  if (wid >= BH * 16 * 4) return;
  const int bh = wid / 64;
  const int r  = wid % 64;
  const int ti = r >> 2;
  const int td = r & 3;

  const int arow = ti * 16 + (lane & 15);
  const int koff = (lane >> 4) * 2;
  const int dcol = td * 16 + (lane & 15);
  const float* A = attn + bh * Nsz * Nsz;
  const float* V = v + bh * Nsz * Dsz;

  v8f c = {};
  for (int jj = 0; jj < Nsz; jj += 4) {
    v2f a, b;
    const float* ap = A + arow * Nsz + jj + koff;
    a.x = ap[0];
    a.y = ap[1];
    b.x = V[(jj + koff) * Dsz + dcol];
    b.y = V[(jj + koff + 1) * Dsz + dcol];
    c = wmma_f32_k4(a, b, c);
  }

  const int ibase = ti * 16 + ((lane >> 4) << 3);
  for (int vv = 0; vv < 8; ++vv)
    out[(bh * Nsz + ibase + vv) * Dsz + dcol] = c[vv];
}

// ---------------------------------------------------------------------------
// Kernel 7: output projection + residual: x = X2 @ Wo + bo + hidden.
// X2 is attn_out viewed flat as [512,768].  waves = 32*48.
// ---------------------------------------------------------------------------
__global__ void __launch_bounds__(128) out_proj_kernel(
    const float* __restrict__ X2, const float* __restrict__ Wo,
    const float* __restrict__ bo, const float* __restrict__ hidden,
    float* __restrict__ x) {
  const int wid  = (blockIdx.x * blockDim.x + threadIdx.x) >> 5;
  const int lane = threadIdx.x & 31;
  if (wid >= 32 * 48) return;
  const int tm = wid / 48;
  const int tn = wid % 48;

  const int arow = tm * 16 + (lane & 15);
  const int koff = (lane >> 4) * 2;
  const int ncol = tn * 16 + (lane & 15);

  v8f c = {};
  for (int kk = 0; kk < Esz; kk += 4) {
    v2f a, b;
    const float* ap = X2 + arow * Esz + kk + koff;
    a.x = ap[0];
    a.y = ap[1];
    b.x = Wo[(kk + koff) * Esz + ncol];
    b.y = Wo[(kk + koff + 1) * Esz + ncol];
    c = wmma_f32_k4(a, b, c);
  }

  const float bb = bo[ncol];
  const int mbase = tm * 16 + ((lane >> 4) << 3);
  for (int vv = 0; vv < 8; ++vv) {
    const int bn = mbase + vv;
    x[bn * Esz + ncol] = c[vv] + bb + hidden[bn * Esz + ncol];
  }
}

// ---------------------------------------------------------------------------
// Kernel 8: LayerNorm over E=768.  One wave per row (512 rows).
// ---------------------------------------------------------------------------
__global__ void __launch_bounds__(256) layernorm_kernel(
    const float* __restrict__ x, const float* __restrict__ g,
    const float* __restrict__ b, float* __restrict__ out) {
  const int wid  = (blockIdx.x * blockDim.x + threadIdx.x) >> 5;
  const int lane = threadIdx.x & 31;
  if (wid >= BN) return;
  const float* r = x + wid * Esz;

  float vals[24];
  float mu = 0.0f;
#pragma unroll
  for (int m = 0; m < 24; ++m) {
    vals[m] = r[lane + m * 32];
    mu += vals[m];
  }
  for (int off = 16; off; off >>= 1) mu += __shfl_xor(mu, off, 32);
  mu *= (1.0f / (float)Esz);

  float var = 0.0f;
#pragma unroll
  for (int m = 0; m < 24; ++m) {
    const float d = vals[m] - mu;
    var += d * d;
  }
  for (int off = 16; off; off >>= 1) var += __shfl_xor(var, off, 32);
  var *= (1.0f / (float)Esz);

  const float rs = rsqrtf(var + 1e-12f);
#pragma unroll
  for (int m = 0; m < 24; ++m) {
    const int e = lane + m * 32;
    out[wid * Esz + e] = (vals[m] - mu) * rs * g[e] + b[e];
  }
}

// ---------------------------------------------------------------------------
// Host launcher
// ---------------------------------------------------------------------------
extern "C" void kernel_launch(void* const* d_in, const int* in_sizes, int n_in,
                              void* d_out, int out_size, void* d_ws, size_t ws_size,
                              hipStream_t stream) {
  (void)in_sizes; (void)n_in; (void)out_size; (void)ws_size;

  const float* hidden = (const float*)d_in[0];
  const float* amask  = (const float*)d_in[1];
  const float* Wq = (const float*)d_in[2];
  const float* bq = (const float*)d_in[3];
  const float* Wk = (const float*)d_in[4];
  const float* bk = (const float*)d_in[5];
  const float* Wv = (const float*)d_in[6];
  const float* bv = (const float*)d_in[7];
  const float* Wo = (const float*)d_in[8];
  const float* bo = (const float*)d_in[9];
  const float* ln_g = (const float*)d_in[10];
  const float* ln_b = (const float*)d_in[11];
  float* out = (float*)d_out;

  // Workspace layout (floats).
  float* w = (float*)d_ws;
  const size_t qkv_elems = (size_t)BH * Nsz * Dsz;   // 393216
  float* q       = w;                                //  1.5 MB
  float* k       = q + qkv_elems;
  float* v       = k + qkv_elems;
  float* attnout = v + qkv_elems;                    //  1.5 MB
  float* x       = attnout + qkv_elems;              //  1.5 MB
  float* q2      = x + qkv_elems;                    //  24 KB
  float* k2      = q2 + (size_t)BH * Nsz;
  float* dist    = k2 + (size_t)BH * Nsz;            //  6 MB

  // 1) QKV projections (WMMA f32).
  qkv_gemm_kernel<<<1152, 128, 0, stream>>>(hidden, Wq, bq, Wk, bk, Wv, bv, q, k, v);
  // 2) Row squared norms.
  rownorm_kernel<<<1536, 256, 0, stream>>>(q, k, q2, k2);
  // 3) Pairwise distances (WMMA f32) + mask.
  cdist_kernel<<<1536, 128, 0, stream>>>(q, k, q2, k2, amask, dist);
  // 4) Floyd-Warshall APSP, 256 KB LDS per workgroup (CDNA5 320 KB/WGP).
  (void)hipFuncSetAttribute((const void*)fw_apsp_kernel,
                            hipFuncAttributeMaxDynamicSharedMemorySize, 256 * 1024);
  fw_apsp_kernel<<<BH, 1024, 256 * 1024, stream>>>(dist);
  // 5) Score transform + softmax (in place).
  score_softmax_kernel<<<768, 256, 0, stream>>>(dist);
  // 6) attn @ V (WMMA f32); flat [B,H,N,D] == reshape [B,N,E].
  av_gemm_kernel<<<384, 128, 0, stream>>>(dist, v, attnout);
  // 7) Output projection + residual (WMMA f32).
  out_proj_kernel<<<384, 128, 0, stream>>>(attnout, Wo, bo, hidden, x);
  // 8) LayerNorm -> d_out.
  layernorm_kernel<<<64, 256, 0, stream>>>(x, ln_g, ln_b, out);
}